// LLLocalClusterCoordinates_5428838662735
// MI455X (gfx1250) — compile-verified
//
#include <hip/hip_runtime.h>
#include <stdint.h>

// LLLocalClusterCoordinates loss for MI455X (gfx1250).
// Memory-bound (~768MB HBM => ~33us floor @ 23.3TB/s). Strategy:
//  - TDM (tensor_load_to_lds) double-buffers 8x64 tiles of dist/nidxs into LDS
//    (TENSORcnt-tracked async DMA, zero VGPR cost, deep prefetch).
//  - wave32: one wave per row, lane l owns K-elements {2l, 2l+1} via b64 loads.
//  - neighbor-count reductions via wave32 ballot+popcount (SALU, no DS traffic);
//    only the two float sums use shfl_xor butterflies.
//  - tidxs gather (4MB table) stays L2-resident on the 192MB L2.
//  - deterministic 2-stage scalar reduction through d_ws (no float atomics).

#define KDIM      64
#define TILE_ROWS 8
#define BLOCK     256        // 8 wave32 waves -> 8 rows per tile
#define GRID      2048

typedef unsigned int v4u __attribute__((ext_vector_type(4)));
typedef int          v4i __attribute__((ext_vector_type(4)));
typedef int          v8i __attribute__((ext_vector_type(8)));

// Build a 2D TDM descriptor (D#) and issue TENSOR_LOAD_TO_LDS.
// Tensor == tile: rows x KDIM of 4-byte elements, row stride KDIM.
// D# layout per CDNA5 ISA ch.8 (groups 0/1; groups 2/3 unused for 2D).
__device__ __forceinline__ void tdm_load_tile_2d(uint64_t gaddr, uint32_t lds_off,
                                                 uint32_t rows) {
  v4u g0;
  g0.x = 1u;                                                    // count=1 (valid D#)
  g0.y = lds_off;                                               // LDS byte address
  g0.z = (uint32_t)gaddr;                                       // global_addr[31:0]
  g0.w = ((uint32_t)(gaddr >> 32) & 0x01FFFFFFu) | (2u << 30);  // addr[56:32] | type=2
  v8i g1;
  g1[0] = 0x00020000;                                  // data_size=2 (4 bytes)
  g1[1] = (int)(((uint32_t)KDIM & 0xFFFFu) << 16);     // tensor_dim0[15:0]=64
  g1[2] = (int)((rows & 0xFFFFu) << 16);               // dim0 hi=0 | tensor_dim1[15:0]=rows
  g1[3] = (int)(((uint32_t)KDIM & 0xFFFFu) << 16);     // dim1 hi=0 | tile_dim0=64
  g1[4] = (int)(rows & 0xFFFFu);                       // tile_dim1=rows, tile_dim2=0
  g1[5] = KDIM;                                        // tensor_dim0_stride[31:0]=64
  g1[6] = 0;                                           // stride hi / dim1_stride lo
  g1[7] = 0;
  v4i z4 = {0, 0, 0, 0};
#if defined(__clang_major__) && (__clang_major__ >= 23)
  v8i z8 = {0, 0, 0, 0, 0, 0, 0, 0};
  __builtin_amdgcn_tensor_load_to_lds(g0, g1, z4, z4, z8, 0);
#else
  __builtin_amdgcn_tensor_load_to_lds(g0, g1, z4, z4, 0);
#endif
}

__device__ __forceinline__ uint32_t lds_offset_of(const void* p) {
  // Flat shared address = {aperture, lds_offset}; low 32 bits are the LDS offset.
  return (uint32_t)(uintptr_t)p;
}

__global__ __launch_bounds__(BLOCK) void oc_loss_kernel(
    const float* __restrict__ dist, const int* __restrict__ nidxs,
    const int* __restrict__ tidxs, float* __restrict__ out,
    float* __restrict__ partials, int N, int numTiles) {
  __shared__ __align__(16) float sDist[2][TILE_ROWS * KDIM];  // 2x2KB
  __shared__ __align__(16) int   sIdx [2][TILE_ROWS * KDIM];  // 2x2KB
  __shared__ float sRed[BLOCK / 32][2];

  const int tid  = threadIdx.x;
  const int wave = tid >> 5;
  const int lane = tid & 31;

  float lsum = 0.0f, psum = 0.0f;

  // Stage first tile (TDM ignores EXEC; issue from exactly one wave's region —
  // other waves skip via execz branch).
  if (tid == 0) {
    int t = (int)blockIdx.x;   // grid is clamped to numTiles by the host
    uint32_t rows = (uint32_t)min(TILE_ROWS, N - t * TILE_ROWS);
    uint64_t boff = (uint64_t)t * (TILE_ROWS * KDIM * 4);
    tdm_load_tile_2d((uint64_t)(uintptr_t)dist  + boff, lds_offset_of(&sDist[0][0]), rows);
    tdm_load_tile_2d((uint64_t)(uintptr_t)nidxs + boff, lds_offset_of(&sIdx [0][0]), rows);
  }

  int cur = 0;
  for (int t = (int)blockIdx.x; t < numTiles; t += (int)gridDim.x) {
    int nt = t + (int)gridDim.x;
    if (tid == 0) {
      if (nt < numTiles) {
        // Prefetch next tile into the other buffer, then wait only for the
        // current tile's two DMAs (TDM ops complete in order per wave).
        uint32_t rows = (uint32_t)min(TILE_ROWS, N - nt * TILE_ROWS);
        uint64_t boff = (uint64_t)nt * (TILE_ROWS * KDIM * 4);
        tdm_load_tile_2d((uint64_t)(uintptr_t)dist  + boff, lds_offset_of(&sDist[cur ^ 1][0]), rows);
        tdm_load_tile_2d((uint64_t)(uintptr_t)nidxs + boff, lds_offset_of(&sIdx [cur ^ 1][0]), rows);
        __builtin_amdgcn_s_wait_tensorcnt(2);
      } else {
        __builtin_amdgcn_s_wait_tensorcnt(0);
      }
    }
    __syncthreads();  // tile data in buf[cur] visible to all waves

    const int r = t * TILE_ROWS + wave;     // one wave32 per row
    if (r < N) {
      float2 d2 = ((const float2*)&sDist[cur][wave * KDIM])[lane];
      int2   i2 = ((const int2  *)&sIdx [cur][wave * KDIM])[lane];

      // Gather cluster ids (tidxs is 4MB -> L2-hot on the 192MB L2).
      const bool vA = (i2.x >= 0);
      const bool vB = (i2.y >= 0);
      int selA = vA ? tidxs[i2.x] : -1;
      int selB = vB ? tidxs[i2.y] : -1;
      int sel0 = __shfl(selA, 0, 32);       // k=0 lives in lane 0

      const bool attA = vA && (selA == sel0);
      const bool attB = vB && (selB == sel0);
      const bool repA = vA && (selA != sel0);
      const bool repB = vB && (selB != sel0);

      // Mask counts for free on wave32: ballot + popcount (SALU), broadcast to
      // all lanes -> no cross-lane float reduction needed for natt/nrep.
      float natt = (float)(__popcll(__ballot(attA)) + __popcll(__ballot(attB)));
      float nrep = (float)(__popcll(__ballot(repA)) + __popcll(__ballot(repB)));

      float nnwA = (selA < 0) ? 0.01f : 1.0f;
      float nnwB = (selB < 0) ? 0.01f : 1.0f;

      const float E = 2.718281828f;
      float att = (attA ? __logf(E * d2.x + 1.0f) : 0.0f)
                + (attB ? __logf(E * d2.y + 1.0f) : 0.0f);
      float rep = (repA ? __expf(-d2.x) * nnwA : 0.0f)
                + (repB ? __expf(-d2.y) * nnwB : 0.0f);

      // Stream the dist passthrough (reference returns inputs[0]); coalesced b64.
      ((float2*)(out + (size_t)r * KDIM))[lane] = d2;

      // wave32 butterfly reduction of the two float row sums.
#pragma unroll
      for (int off = 16; off > 0; off >>= 1) {
        att += __shfl_xor(att, off, 32);
        rep += __shfl_xor(rep, off, 32);
      }

      if (lane == 0) {
        float pn = (tidxs[r] >= 0) ? 1.0f : 0.0f;
        float attloss = (natt == 0.0f) ? 0.0f : (pn * att) / natt;  // div_no_nan
        float reploss = (nrep == 0.0f) ? 0.0f : (pn * rep) / nrep;
        lsum += pn * (attloss + reploss);
        psum += pn;
      }
    }
    __syncthreads();  // all reads of buf[cur] done before it is DMA-overwritten
    cur ^= 1;
  }

  if (lane == 0) { sRed[wave][0] = lsum; sRed[wave][1] = psum; }
  __syncthreads();
  if (tid == 0) {
    float L = 0.0f, P = 0.0f;
#pragma unroll
    for (int w = 0; w < BLOCK / 32; ++w) { L += sRed[w][0]; P += sRed[w][1]; }
    partials[2 * blockIdx.x]     = L;   // every block writes -> d_ws fully initialized
    partials[2 * blockIdx.x + 1] = P;
  }
}

__global__ __launch_bounds__(BLOCK) void oc_loss_reduce(
    const float* __restrict__ partials, int nPart, float* __restrict__ lossOut) {
  __shared__ float sL[BLOCK], sP[BLOCK];
  float L = 0.0f, P = 0.0f;
  for (int i = threadIdx.x; i < nPart; i += BLOCK) {
    L += partials[2 * i];
    P += partials[2 * i + 1];
  }
  sL[threadIdx.x] = L; sP[threadIdx.x] = P;
  __syncthreads();
  for (int s = BLOCK / 2; s > 0; s >>= 1) {
    if (threadIdx.x < s) {
      sL[threadIdx.x] += sL[threadIdx.x + s];
      sP[threadIdx.x] += sP[threadIdx.x + s];
    }
    __syncthreads();
  }
  if (threadIdx.x == 0)
    lossOut[0] = (sP[0] == 0.0f) ? 0.0f : sL[0] / sP[0];  // final div_no_nan
}

extern "C" void kernel_launch(void* const* d_in, const int* in_sizes, int n_in,
                              void* d_out, int out_size, void* d_ws, size_t ws_size,
                              hipStream_t stream) {
  const float* dist  = (const float*)d_in[0];   // [N,K] f32
  const int*   nidxs = (const int*)  d_in[1];   // [N,K] i32
  const int*   tidxs = (const int*)  d_in[2];   // [N,1] i32
  // d_in[3] (specweight) is dead code in the reference loss.
  float* out = (float*)d_out;                   // [N*K] dist copy + [1] loss

  const int N = in_sizes[2];                    // tidxs flat size == N
  const int numTiles = (N + TILE_ROWS - 1) / TILE_ROWS;
  int grid = GRID;
  if (grid > numTiles) grid = numTiles;

  float* partials = (float*)d_ws;               // grid pairs, rewritten every call

  oc_loss_kernel<<<grid, BLOCK, 0, stream>>>(dist, nidxs, tidxs, out, partials,
                                             N, numTiles);
  oc_loss_reduce<<<1, BLOCK, 0, stream>>>(partials, grid, out + (size_t)N * KDIM);
}